// GroundTruthGroupingTeacher_91087666413582
// MI455X (gfx1250) — compile-verified
//
#include <hip/hip_runtime.h>

// Problem geometry (from reference)
#define R        7
#define KS       15          // kernel side (2R+1)
#define KK       225         // KS*KS
#define OH       256         // output H (512/2)
#define OW       256         // output W
#define SRC_W    512
#define NB       8

// Tiling
#define TX       64
#define TY       32
#define LDS_H    (TY + 2*R)  // 46
#define LDS_W    (TX + 2*R)  // 78
#define LDS_P    80          // padded pitch (dwords)

typedef float v4f __attribute__((ext_vector_type(4)));

// Address-space-qualified pointer types for the async-to-LDS builtin
// (toolchain declares the params as int* in AS1 / AS3)
typedef __attribute__((address_space(1))) int as1_int;
typedef __attribute__((address_space(3))) int as3_int;

#ifndef __has_builtin
#define __has_builtin(x) 0
#endif

#if __has_builtin(__builtin_amdgcn_global_load_async_to_lds_b32)
#define USE_ASYNC_LDS 1
#endif

__global__ __launch_bounds__(256)
void adj_local_eq_kernel(const float* __restrict__ src, float* __restrict__ out)
{
    __shared__ float lds[LDS_H * LDS_P];

    const int tid = threadIdx.x;
    const int x0  = blockIdx.x * TX;   // tile origin x (output coords)
    const int y0  = blockIdx.y * TY;   // tile origin y
    const int b   = blockIdx.z;        // batch

    const float* __restrict__ srcb = src + (size_t)b * SRC_W * SRC_W;

    // ---------------- stage subsampled tile (+halo) into LDS, OOB -> 0 ----------------
#ifdef USE_ASYNC_LDS
    // zero-fill (covers padded columns + OOB cells), then async-copy in-bounds cells
    for (int e = tid; e < LDS_H * LDS_P; e += 256) lds[e] = 0.0f;
    __syncthreads();                               // DS writes complete before async writes
    for (int e = tid; e < LDS_H * LDS_W; e += 256) {
        const int r  = e / LDS_W;
        const int c  = e - r * LDS_W;
        const int gy = y0 - R + r;                 // subsampled-image coords
        const int gx = x0 - R + c;
        if ((unsigned)gy < OH && (unsigned)gx < OW) {
            const float* gp = srcb + (size_t)(2 * gy) * SRC_W + (2 * gx);
            __builtin_amdgcn_global_load_async_to_lds_b32(
                (as1_int*)gp,
                (as3_int*)&lds[r * LDS_P + c],
                /*offset=*/0, /*cpol=*/0);
        }
    }
#if __has_builtin(__builtin_amdgcn_s_wait_asynccnt)
    __builtin_amdgcn_s_wait_asynccnt(0);
#else
    asm volatile("s_wait_asynccnt 0" ::: "memory");
#endif
    __syncthreads();
#else
    for (int e = tid; e < LDS_H * LDS_W; e += 256) {
        const int r  = e / LDS_W;
        const int c  = e - r * LDS_W;
        const int gy = y0 - R + r;
        const int gx = x0 - R + c;
        float v = 0.0f;
        if ((unsigned)gy < OH && (unsigned)gx < OW)
            v = srcb[(size_t)(2 * gy) * SRC_W + (2 * gx)];
        lds[r * LDS_P + c] = v;
    }
    __syncthreads();
#endif

    // ---------------- per-thread output assignment ----------------
    // lanes 0..15 of each half cover one contiguous 256B store line
    const int tx4 = tid & 15;       // float4 column
    const int ty  = tid >> 4;       // 0..15 ; thread owns rows ty and ty+16
    const int xl  = tx4 * 4;        // local x of first owned output

    // center labels (fixed across all 225 offsets)
    float c0[4], c1[4];
#pragma unroll
    for (int i = 0; i < 4; ++i) {
        c0[i] = lds[(R + ty     ) * LDS_P + (R + xl + i)];
        c1[i] = lds[(R + ty + 16) * LDS_P + (R + xl + i)];
    }

    float* __restrict__ o0 =
        out + (((size_t)b * KK) * OH + (size_t)(y0 + ty)) * OW + (x0 + xl);
    float* __restrict__ o1 = o0 + (size_t)16 * OW;

    for (int dy = 0; dy < KS; ++dy) {
        // 18-wide sliding windows (4 outputs + 14 slide) held in VGPRs
        float w0[18], w1[18];
#pragma unroll
        for (int j = 0; j < 18; ++j) {
            w0[j] = lds[(ty      + dy) * LDS_P + xl + j];
            w1[j] = lds[(ty + 16 + dy) * LDS_P + xl + j];
        }
#pragma unroll
        for (int dx = 0; dx < KS; ++dx) {
            const size_t koff = (size_t)(dy * KS + dx) * (OH * OW);
            v4f r0, r1;
            r0.x = (c0[0] == w0[dx + 0]) ? 1.0f : 0.0f;
            r0.y = (c0[1] == w0[dx + 1]) ? 1.0f : 0.0f;
            r0.z = (c0[2] == w0[dx + 2]) ? 1.0f : 0.0f;
            r0.w = (c0[3] == w0[dx + 3]) ? 1.0f : 0.0f;
            r1.x = (c1[0] == w1[dx + 0]) ? 1.0f : 0.0f;
            r1.y = (c1[1] == w1[dx + 1]) ? 1.0f : 0.0f;
            r1.z = (c1[2] == w1[dx + 2]) ? 1.0f : 0.0f;
            r1.w = (c1[3] == w1[dx + 3]) ? 1.0f : 0.0f;
            // streaming output (472MB total, never re-read): non-temporal b128 stores
            __builtin_nontemporal_store(r0, (v4f*)(o0 + koff));
            __builtin_nontemporal_store(r1, (v4f*)(o1 + koff));
        }
    }
}

extern "C" void kernel_launch(void* const* d_in, const int* in_sizes, int n_in,
                              void* d_out, int out_size, void* d_ws, size_t ws_size,
                              hipStream_t stream)
{
    (void)in_sizes; (void)n_in; (void)d_ws; (void)ws_size; (void)out_size;
    const float* src = (const float*)d_in[0];
    float*       out = (float*)d_out;
    dim3 grid(OW / TX, OH / TY, NB);   // (4, 8, 8) = 256 blocks
    adj_local_eq_kernel<<<grid, 256, 0, stream>>>(src, out);
}